// CausalRevIN_4922032521622
// MI455X (gfx1250) — compile-verified
//
#include <hip/hip_runtime.h>
#include <hip/hip_bf16.h>
#include <stdint.h>

// Problem dims (fixed by reference)
#define B_    16
#define T_    8192
#define C_    128
#define NCH   64            // chunks along T (parallel scan decomposition)
#define CT    (T_ / NCH)    // 128 timesteps per chunk
#define TT    4             // timesteps per LDS tile
#define NTILE (CT / TT)     // 32 tiles per chunk
#define LANES 32
#define WPB   4             // waves per 128-thread block
#define NBUF  4             // ring of 4 tile buffers -> issue 3 tiles ahead
#define TILE_BYTES (TT * 512)                   // one array, one buffer: 2 KB
#define WAVE_LDS   (NBUF * 2 * TILE_BYTES)      // 4 bufs x {x,mask}: 16 KB
#define BLOCK_LDS  (WPB * WAVE_LDS)             // 64 KB

typedef float v4f __attribute__((ext_vector_type(4)));

// One wave handles one (b, chunk); each lane owns 4 consecutive channels so
// every async b128 load is a contiguous 512B/wave transfer. Tiles of TT
// timesteps of x and mask stream through a 4-deep LDS ring via gfx1250
// async global->LDS loads; the serial scan consumes them with ds_load_b128.
template <int PASS>
__global__ __launch_bounds__(128) void crevin_pass(
    const float* __restrict__ x, const float* __restrict__ mask,
    float* __restrict__ out,
    v4f* __restrict__ pS, v4f* __restrict__ pN, v4f* __restrict__ pD)
{
    __shared__ char lds[BLOCK_LDS];

    const int tid   = threadIdx.x;
    const int lane  = tid & 31;
    const int wave  = tid >> 5;
    const int gw    = blockIdx.x * WPB + wave;   // global wave id = (b, chunk)
    const int chunk = gw & (NCH - 1);
    const int b     = gw >> 6;

    // element index of this lane's float4 at t = chunk*CT
    const size_t base = ((size_t)b * T_ + (size_t)chunk * CT) * C_ + lane * 4;
    const uint64_t xga = (uint64_t)(uintptr_t)(x + base);
    const uint64_t mga = (uint64_t)(uintptr_t)(mask + base);

    char* ldsWave = lds + wave * WAVE_LDS;

    // Issue one tile (TT timesteps of x and mask) of async b128 loads.
    auto issue = [&](int tile) {
        char* bufBase = ldsWave + (tile & (NBUF - 1)) * (2 * TILE_BYTES);
#pragma unroll
        for (int tt = 0; tt < TT; ++tt) {
            const uint64_t byteOff = (uint64_t)(tile * TT + tt) * (C_ * 4);
            uint64_t gx = xga + byteOff;
            uint64_t gm = mga + byteOff;
            unsigned lx = (unsigned)(uintptr_t)(bufBase + tt * 512 + lane * 16);
            unsigned lm = (unsigned)(uintptr_t)(bufBase + TILE_BYTES + tt * 512 + lane * 16);
            if constexpr (PASS == 2) {
                // last reader of x/mask: last-use policy frees L2 as we stream
                asm volatile("global_load_async_to_lds_b128 %0, %1, off th:TH_LOAD_LU"
                             :: "v"(lx), "v"(gx) : "memory");
                asm volatile("global_load_async_to_lds_b128 %0, %1, off th:TH_LOAD_LU"
                             :: "v"(lm), "v"(gm) : "memory");
            } else {
                asm volatile("global_load_async_to_lds_b128 %0, %1, off"
                             :: "v"(lx), "v"(gx) : "memory");
                asm volatile("global_load_async_to_lds_b128 %0, %1, off"
                             :: "v"(lm), "v"(gm) : "memory");
            }
        }
    };

    const int pBase = (b * NCH + chunk) * LANES + lane;

    v4f S = 0.0f, N = 0.0f, D = 0.0f;
    if constexpr (PASS >= 1) { S = pS[pBase]; N = pN[pBase]; }  // exclusive prefixes
    if constexpr (PASS == 2) { D = pD[pBase]; }

    // Consume one tile out of the LDS ring.
    auto consume = [&](int tile) {
        char* bufBase = ldsWave + (tile & (NBUF - 1)) * (2 * TILE_BYTES);
#pragma unroll
        for (int tt = 0; tt < TT; ++tt) {
            v4f xv = *(const v4f*)(bufBase + tt * 512 + lane * 16);
            v4f mv = *(const v4f*)(bufBase + TILE_BYTES + tt * 512 + lane * 16);
            v4f nm = 1.0f - mv;
            S += xv;
            N += nm;
            if constexpr (PASS >= 1) {
                v4f o;
#pragma unroll
                for (int i = 0; i < 4; ++i) {
                    float n    = (N[i] == 0.0f) ? 1.0f : N[i];
                    float rn   = 1.0f / n;          // one divide, reused
                    float mean = S[i] * rn;
                    float dv   = (xv[i] - mean) * nm[i];
                    D[i] += dv * dv;
                    if constexpr (PASS == 2) {
                        float sd = __builtin_sqrtf(D[i] * rn);
                        sd = (sd > 1e-5f) ? sd : 1.0f;
                        float r = (xv[i] - mean) / sd;
                        r = fminf(fmaxf(r, -100.0f), 100.0f);
                        o[i] = r;
                    }
                }
                if constexpr (PASS == 2) {
                    const size_t t = (size_t)(tile * TT + tt);
                    __builtin_nontemporal_store(o, (v4f*)(out + base + t * C_));
                }
            }
        }
    };

    // Prime the ring 3 tiles deep (24 async ops in flight per wave).
    issue(0); issue(1); issue(2);
    int tile = 0;
    for (; tile < NTILE - 3; ++tile) {
        issue(tile + 3);                                 // 32 in flight
        asm volatile("s_wait_asynccnt 24" ::: "memory"); // oldest tile landed
        consume(tile);
    }
    asm volatile("s_wait_asynccnt 16" ::: "memory"); consume(NTILE - 3);
    asm volatile("s_wait_asynccnt 8"  ::: "memory"); consume(NTILE - 2);
    asm volatile("s_wait_asynccnt 0"  ::: "memory"); consume(NTILE - 1);

    if constexpr (PASS == 0) { pS[pBase] = S; pN[pBase] = N; }
    if constexpr (PASS == 1) { pD[pBase] = D; }
}

// Serial exclusive prefix over the chunk dimension for one (or two) partial
// arrays laid out as [b][chunk][lane][4] floats. One thread per scalar
// channel: 16*32*4 = 2048 threads. Tiny kernel (workspace lives in L2).
__global__ __launch_bounds__(256) void crevin_prefix(float* __restrict__ a,
                                                     float* __restrict__ b2)
{
    const int p    = blockIdx.x * blockDim.x + threadIdx.x;  // 0..2047
    const int comp = p & 3;
    const int lane = (p >> 2) & 31;
    const int bb   = p >> 7;
    float ra = 0.0f, rb = 0.0f;
    for (int ch = 0; ch < NCH; ++ch) {
        const size_t idx =
            ((size_t)((bb * NCH + ch) * LANES + lane) << 2) + comp;
        float va = a[idx]; a[idx] = ra; ra += va;
        if (b2) { float vb = b2[idx]; b2[idx] = rb; rb += vb; }
    }
}

extern "C" void kernel_launch(void* const* d_in, const int* in_sizes, int n_in,
                              void* d_out, int out_size, void* d_ws, size_t ws_size,
                              hipStream_t stream) {
    (void)in_sizes; (void)n_in; (void)out_size; (void)ws_size;
    const float* x    = (const float*)d_in[0];
    const float* mask = (const float*)d_in[1];
    float* out = (float*)d_out;
    float* ws  = (float*)d_ws;

    // Workspace: three partial arrays of B*NCH*32 float4 = 512 KB each.
    v4f* pS = (v4f*)ws;
    v4f* pN = (v4f*)(ws + 131072);
    v4f* pD = (v4f*)(ws + 262144);

    const int blocks = (B_ * NCH) / WPB;  // 256 blocks of 128 threads

    // Sweep 1: per-chunk sums of x and (1-mask). HBM-bound; parks input in L2.
    crevin_pass<0><<<blocks, 128, 0, stream>>>(x, mask, out, pS, pN, pD);
    // Exclusive prefix of (S, N) over chunks.
    crevin_prefix<<<8, 256, 0, stream>>>((float*)pS, (float*)pN);
    // Sweep 2: per-chunk sums of ((x-mean)*nm)^2 using prefix offsets. L2-resident.
    crevin_pass<1><<<blocks, 128, 0, stream>>>(x, mask, out, pS, pN, pD);
    // Exclusive prefix of D over chunks.
    crevin_prefix<<<8, 256, 0, stream>>>((float*)pD, nullptr);
    // Sweep 3: final normalize + clip; LU loads drain L2, NT b128 stores.
    crevin_pass<2><<<blocks, 128, 0, stream>>>(x, mask, out, pS, pN, pD);
}